// season_block_46428596469890
// MI455X (gfx1250) — compile-verified
//
#include <hip/hip_runtime.h>

// out[row, t] = 1024*x[row,t] - S0[row] - (-1)^t * S1[row]
//   S0 = sum_t x,  S1 = sum_t x*(-1)^t
// (closed form of the reference: IDFT band reconstruction minus DC & Nyquist,
//  unnormalized FFT -> factor T=1024). Memory-bound: 64 MB total -> ~2.8us @ 23.3 TB/s.
// Row reductions done exactly in f32 via V_WMMA_F32_16X16X4_F32 against a +-1 B matrix.

typedef __attribute__((ext_vector_type(2))) float v2f;
typedef __attribute__((ext_vector_type(8))) float v8f;

#define T_LEN 1024
#define ROWS_PER_BLOCK 16
#define LDS_PAD 4                         // 16B pad -> conflict-free A-fragment ds_load_b64
#define LDS_STRIDE (T_LEN + LDS_PAD)      // 1028 floats per row
#define THREADS 256
#define WAVES (THREADS / 32)              // 8 waves (wave32)
#define KCHUNK (T_LEN / WAVES)            // 128 columns per wave
#define V4_PER_THREAD (ROWS_PER_BLOCK * T_LEN / 4 / THREADS)  // 16

__global__ __launch_bounds__(THREADS)
void season_band_kernel(const float* __restrict__ x, float* __restrict__ out) {
    __shared__ float tile[ROWS_PER_BLOCK * LDS_STRIDE];
    __shared__ float sS0[ROWS_PER_BLOCK];
    __shared__ float sS1[ROWS_PER_BLOCK];

    const int tid  = threadIdx.x;
    const int lane = tid & 31;
    const int wave = tid >> 5;
    const long long row0 = (long long)blockIdx.x * ROWS_PER_BLOCK;
    const float* __restrict__ src = x  + row0 * T_LEN;   // tile is one contiguous 64KB run
    float* __restrict__       dst = out + row0 * T_LEN;

    if (tid < ROWS_PER_BLOCK)            sS0[tid] = 0.0f;
    else if (tid < 2 * ROWS_PER_BLOCK)   sS1[tid - ROWS_PER_BLOCK] = 0.0f;

    // ---- Stage tile: coalesced global_load_b128 -> padded LDS rows ----
    #pragma unroll
    for (int i = 0; i < V4_PER_THREAD; ++i) {
        const int v    = tid + i * THREADS;        // float4 index within tile
        const int flat = v * 4;
        const int r    = flat >> 10;               // /1024
        const int col  = flat & (T_LEN - 1);
        const float4 d = ((const float4*)src)[v];
        *(float4*)&tile[r * LDS_STRIDE + col] = d;
    }
    __syncthreads();

    // ---- Per-wave exact-f32 reduction via chained V_WMMA_F32_16X16X4_F32 ----
    // A (16x4 f32): lane m(0..15) holds K={0,1}; lanes 16..31 hold K={2,3}.
    // B (4x16 f32) constant across steps: col n<8 weight 1 (S0),
    //   col n>=8 weight (-1)^k (S1); k parity == step-local k parity (steps of 4).
    //   -> B vgpr0 (even k rows) = 1.0 all lanes; vgpr1 (odd k rows) = +-1 by lane%16<8.
    const int m    = lane & 15;
    const int half = lane >> 4;
    v2f b;
    b.x = 1.0f;
    b.y = (m < 8) ? 1.0f : -1.0f;
    v8f acc = {};
    const int kbase = wave * KCHUNK + 2 * half;
    #pragma unroll
    for (int s = 0; s < KCHUNK / 4; ++s) {     // 32 chained WMMAs per wave
        v2f a = *(const v2f*)&tile[m * LDS_STRIDE + kbase + 4 * s];
        acc = __builtin_amdgcn_wmma_f32_16x16x4_f32(
            /*neg_a=*/false, a, /*neg_b=*/false, b,
            /*c_mod=*/(short)0, acc, /*reuse_a=*/false, /*reuse_b=*/false);
    }
    // D layout: lane l -> N = l%16, rows M = 8*(l/16)+r in acc[r].
    // Column N=0 carries S0 partials, column N=8 carries S1 partials.
    if (m == 0 || m == 8) {
        float* sdst = (m == 0) ? sS0 : sS1;
        #pragma unroll
        for (int r = 0; r < 8; ++r)
            atomicAdd(&sdst[8 * half + r], acc[r]);   // ds_add_f32 across 8 waves
    }
    __syncthreads();

    // ---- Emit: out = 1024*x - S0 - (-1)^t * S1 (x re-read from LDS, not HBM) ----
    #pragma unroll
    for (int i = 0; i < V4_PER_THREAD; ++i) {
        const int v    = tid + i * THREADS;
        const int flat = v * 4;
        const int r    = flat >> 10;
        const int col  = flat & (T_LEN - 1);       // multiple of 4 -> parity +,-,+,-
        const float4 xv = *(const float4*)&tile[r * LDS_STRIDE + col];
        const float s0 = sS0[r];
        const float s1 = sS1[r];
        float4 o;
        o.x = 1024.0f * xv.x - s0 - s1;
        o.y = 1024.0f * xv.y - s0 + s1;
        o.z = 1024.0f * xv.z - s0 - s1;
        o.w = 1024.0f * xv.w - s0 + s1;
        ((float4*)dst)[v] = o;
    }
}

extern "C" void kernel_launch(void* const* d_in, const int* in_sizes, int n_in,
                              void* d_out, int out_size, void* d_ws, size_t ws_size,
                              hipStream_t stream) {
    (void)n_in; (void)d_ws; (void)ws_size; (void)out_size;
    const float* x = (const float*)d_in[0];
    float* out     = (float*)d_out;
    const int nrows  = in_sizes[0] / T_LEN;          // 64*128 = 8192
    const int blocks = nrows / ROWS_PER_BLOCK;       // 512
    season_band_kernel<<<blocks, THREADS, 0, stream>>>(x, out);
}